// DUAL_78237124264373
// MI455X (gfx1250) — compile-verified
//
#include <hip/hip_runtime.h>
#include <math.h>

typedef float v2f __attribute__((ext_vector_type(2)));
typedef float v8f __attribute__((ext_vector_type(8)));

#define NZ       768
#define NHALF    384
#define DIM      64
#define NPER     200
#define NSTAT    201     // col 0 = identity (U), cols 1..200 = perms (U_b)
#define NCOL     208     // 13 * 16 padded columns
#define ONES_COL 201     // sigma column of all ones -> V[:,201] = row sums r

// workspace layout in floats
#define OFF_ZC   0
#define OFF_SQ   (OFF_ZC + NZ*DIM)        // 49152
#define OFF_SIG  (OFF_SQ + NZ)            // 49920   SigT: NCOL x NZ (row = sigma column)
#define OFF_S    (OFF_SIG + NCOL*NZ)      // 209664  S: 4 x NZ x NZ
#define OFF_V    (OFF_S + 4*NZ*NZ)        // 2568960 V: 4 x NZ x NCOL
// total = 3,207,936 floats = 12.8 MB

// -------- stage 0: pack Z = concat(X,Y), squared norms --------
__global__ void mmd_build_z_sq(const float* __restrict__ X, const float* __restrict__ Y,
                               float* __restrict__ Zc, float* __restrict__ sq) {
  int u = blockIdx.x * blockDim.x + threadIdx.x;
  if (u >= NZ) return;
  const float* src = (u < NHALF) ? (X + (size_t)u * DIM) : (Y + (size_t)(u - NHALF) * DIM);
  float acc = 0.0f;
#pragma unroll 8
  for (int d = 0; d < DIM; ++d) {
    float v = src[d];
    Zc[(size_t)u * DIM + d] = v;
    acc += v * v;
  }
  sq[u] = acc;
}

// -------- stage 0b: constant sigma rows (identity, ones, zero-pad) --------
__global__ void mmd_sigma_base(float* __restrict__ SigT) {
  int idx = blockIdx.x * blockDim.x + threadIdx.x;
  if (idx >= NCOL * NZ) return;
  int row = idx / NZ, v = idx % NZ;
  if (row == 0)             SigT[idx] = (v < NHALF) ? 1.0f : -1.0f;   // unpermuted
  else if (row == ONES_COL) SigT[idx] = 1.0f;                          // row-sum col
  else if (row > ONES_COL)  SigT[idx] = 0.0f;                          // padding
  // rows 1..200 fully written by scatter kernel
}

// -------- stage 0c: scatter permutation membership into sigma rows --------
__global__ void mmd_sigma_scatter(const int* __restrict__ perms, float* __restrict__ SigT) {
  int idx = blockIdx.x * blockDim.x + threadIdx.x;
  if (idx >= NPER * NZ) return;
  int p = idx / NZ, i = idx % NZ;
  int v = perms[idx];
  SigT[(size_t)(1 + p) * NZ + v] = (i < NHALF) ? 1.0f : -1.0f;
}

// -------- stage 1: Gram via f32 WMMA, then 4 kernel matrices --------
// one wave per 16x16 tile of the 768x768 matrix
__global__ void mmd_gram_s(const float* __restrict__ Zc, const float* __restrict__ sq,
                           const float* __restrict__ bw, float* __restrict__ S) {
  const int m0 = blockIdx.x * 16, n0 = blockIdx.y * 16;
  const int lane = threadIdx.x;
  const int r16 = lane & 15, hi = lane >> 4;

  const float* arow = Zc + (size_t)(m0 + r16) * DIM;
  const float* brow = Zc + (size_t)(n0 + r16) * DIM;

  v8f c = {};
#pragma unroll
  for (int k0 = 0; k0 < DIM; k0 += 4) {
    int kk = k0 + 2 * hi;
    v2f a = *(const v2f*)(arow + kk);   // A[m][kk..kk+1]
    v2f b = *(const v2f*)(brow + kk);   // B[kk..kk+1][n] = Z[n][kk..kk+1]
    c = __builtin_amdgcn_wmma_f32_16x16x4_f32(false, a, false, b, (short)0, c, false, false);
  }

  const float b0 = bw[0], b1 = bw[1], b2 = bw[2], b3 = bw[3];
  const float ig0 = 1.0f / (b0 * b0);
  const float il1 = 1.0f / b1;
  const float ig2 = 1.0f / (b2 * b2);
  const float il3 = 1.0f / b3;

  float* S0 = S;
  float* S1 = S + (size_t)1 * NZ * NZ;
  float* S2 = S + (size_t)2 * NZ * NZ;
  float* S3 = S + (size_t)3 * NZ * NZ;

  const int n = n0 + r16;
  const float sqn = sq[n];
#pragma unroll
  for (int r = 0; r < 8; ++r) {
    int m = m0 + r + 8 * hi;
    float g  = c[r];
    float d2 = fmaxf(sq[m] + sqn - 2.0f * g, 0.0f);
    float dist = sqrtf(d2 + 1e-12f);
    float dd = dist * dist;
    size_t o = (size_t)m * NZ + n;
    S0[o] = expf(-dd  * ig0);
    S1[o] = expf(-dist * il1);
    S2[o] = expf(-dd  * ig2);
    S3[o] = expf(-dist * il3);
  }
}

// -------- stage 2: V = S x Sigma via f32 WMMA (K = 768) --------
__global__ void mmd_gemm_v(const float* __restrict__ S, const float* __restrict__ SigT,
                           float* __restrict__ V) {
  const int m0 = blockIdx.x * 16;          // 48 tiles
  const int n0 = blockIdx.y * 16;          // 13 tiles
  const int k  = blockIdx.z;               // 4 kernels
  const int lane = threadIdx.x;
  const int r16 = lane & 15, hi = lane >> 4;

  const float* Sk = S + (size_t)k * NZ * NZ;
  float*       Vk = V + (size_t)k * NZ * NCOL;
  const float* arow = Sk + (size_t)(m0 + r16) * NZ;      // S row (m)
  const float* brow = SigT + (size_t)(n0 + r16) * NZ;    // sigma column (n) stored as row

  v8f c = {};
#pragma unroll 4
  for (int k0 = 0; k0 < NZ; k0 += 4) {
    int kk = k0 + 2 * hi;
    v2f a = *(const v2f*)(arow + kk);
    v2f b = *(const v2f*)(brow + kk);
    c = __builtin_amdgcn_wmma_f32_16x16x4_f32(false, a, false, b, (short)0, c, false, false);
  }
#pragma unroll
  for (int r = 0; r < 8; ++r)
    Vk[(size_t)(m0 + r + 8 * hi) * NCOL + n0 + r16] = c[r];
}

// -------- stage 3: per-(kernel, statistic) reduction + corrections --------
__device__ __forceinline__ float block_reduce9(float val, float* smem, int tid) {
#pragma unroll
  for (int off = 16; off > 0; off >>= 1) val += __shfl_down(val, off, 32);
  int wid = tid >> 5, lid = tid & 31;
  if (lid == 0) smem[wid] = val;
  __syncthreads();
  if (wid == 0) {
    val = (lid < 8) ? smem[lid] : 0.0f;
#pragma unroll
    for (int off = 4; off > 0; off >>= 1) val += __shfl_down(val, off, 32);
  }
  __syncthreads();
  return val;  // valid on tid 0
}

__global__ void mmd_stats(const float* __restrict__ S, const float* __restrict__ SigT,
                          const float* __restrict__ V, const int* __restrict__ perms,
                          float* __restrict__ out) {
  __shared__ float smem[8];
  const int bid = blockIdx.x;            // 0..803
  const int k = bid / NSTAT, p = bid % NSTAT;
  const float* Sk = S + (size_t)k * NZ * NZ;
  const float* Vk = V + (size_t)k * NZ * NCOL;
  const float* sg = SigT + (size_t)p * NZ;
  const int tid = threadIdx.x;

  float q = 0, cs = 0, Ts = 0, dA = 0, dT = 0;
  for (int u = tid; u < NZ; u += 256) {
    float vv = Vk[(size_t)u * NCOL + p];
    float s  = sg[u];
    q  += s * vv;                               // sigma' S sigma contribution
    cs += vv;                                   // colsum -> r' sigma
    Ts += Vk[(size_t)u * NCOL + ONES_COL];      // T = sum(S)
    float du = Sk[(size_t)u * NZ + u];          // diag
    dT += du;
    if (s > 0.0f) dA += du;
  }

  float Zaa = 0, Zab = 0, Zbb = 0, P = 0;
  for (int j = tid; j < NHALF; j += 256) {
    float zj = Sk[(size_t)j * NZ + NHALF + j];  // the zeroed entries' values
    bool aj = sg[j] > 0.0f;
    bool bj = sg[NHALF + j] > 0.0f;
    if (aj) { if (bj) Zaa += zj; else Zab += zj; }
    else if (!bj) Zbb += zj;
    if (p > 0) {                                // paired term K[perm_X[i], perm_Y[i]]
      int ai = perms[(size_t)(p - 1) * NZ + j];
      int bi = perms[(size_t)(p - 1) * NZ + NHALF + j];
      float kv = Sk[(size_t)ai * NZ + bi];
      if (ai < NHALF && bi == ai + NHALF) kv = 0.0f;   // zeroed entry of K
      P += kv;
    }
  }

  q   = block_reduce9(q,   smem, tid);
  cs  = block_reduce9(cs,  smem, tid);
  Ts  = block_reduce9(Ts,  smem, tid);
  dA  = block_reduce9(dA,  smem, tid);
  dT  = block_reduce9(dT,  smem, tid);
  Zaa = block_reduce9(Zaa, smem, tid);
  Zab = block_reduce9(Zab, smem, tid);
  Zbb = block_reduce9(Zbb, smem, tid);
  P   = block_reduce9(P,   smem, tid);

  if (tid == 0) {
    float T   = Ts;
    float Qab = 0.25f * (T - q);
    float tA  = 0.5f  * (T + cs);
    float Qaa = tA - Qab;
    float Qbb = 0.5f * (T - cs) - Qab;
    float sXX = Qaa - Zaa - dA;
    float sYY = Qbb - Zbb - (dT - dA);
    float sXY = Qab - Zab - P;                  // P == 0 for the identity column
    const float c1 = 1.0f / (384.0f * 383.0f);
    const float c2 = 2.0f / (384.0f * 384.0f);
    out[(size_t)k * NSTAT + p] = sXX * c1 + sYY * c1 - sXY * c2;
  }
}

extern "C" void kernel_launch(void* const* d_in, const int* in_sizes, int n_in,
                              void* d_out, int out_size, void* d_ws, size_t ws_size,
                              hipStream_t stream) {
  const float* X     = (const float*)d_in[0];   // 384x64
  const float* Y     = (const float*)d_in[1];   // 384x64
  const float* bw    = (const float*)d_in[2];   // 4
  const int*   perms = (const int*)d_in[3];     // 200x768
  float* out = (float*)d_out;                   // 4x201

  float* ws   = (float*)d_ws;
  float* Zc   = ws + OFF_ZC;
  float* sq   = ws + OFF_SQ;
  float* SigT = ws + OFF_SIG;
  float* S    = ws + OFF_S;
  float* V    = ws + OFF_V;

  mmd_build_z_sq<<<(NZ + 255) / 256, 256, 0, stream>>>(X, Y, Zc, sq);
  mmd_sigma_base<<<(NCOL * NZ + 255) / 256, 256, 0, stream>>>(SigT);
  mmd_sigma_scatter<<<(NPER * NZ + 255) / 256, 256, 0, stream>>>(perms, SigT);
  mmd_gram_s<<<dim3(NZ / 16, NZ / 16), 32, 0, stream>>>(Zc, sq, bw, S);
  mmd_gemm_v<<<dim3(NZ / 16, NCOL / 16, 4), 32, 0, stream>>>(S, SigT, V);
  mmd_stats<<<4 * NSTAT, 256, 0, stream>>>(S, SigT, V, perms, out);
}